// DotProductPredictor_68066641707579
// MI455X (gfx1250) — compile-verified
//
#include <hip/hip_runtime.h>
#include <hip/hip_bf16.h>

// Per-edge dot product: score[e] = dot(h[src[e]], h[dst[e]]), D_FEAT = 64, fp32.
//
// Memory-bound gather kernel: h (25.6 MB) is L2-resident on MI455X (192 MB L2),
// so we optimize for wide L2-hit gathers (b128 loads) and use the CDNA5 matrix
// pipe (V_WMMA_F32_16X16X4_F32 with an all-ones B) as a 16-edge cross-lane
// reduction engine instead of a shuffle tree.

typedef __attribute__((ext_vector_type(2))) float v2f;
typedef __attribute__((ext_vector_type(4))) float v4f;
typedef __attribute__((ext_vector_type(8))) float v8f;

#define D_FEAT 64
#define EDGES_PER_WAVE 16
#define BLOCK_THREADS 256
#define EDGES_PER_BLOCK ((BLOCK_THREADS / 32) * EDGES_PER_WAVE)

__global__ __launch_bounds__(BLOCK_THREADS) void edge_dot_wmma_kernel(
    const float* __restrict__ h,
    const int* __restrict__ src,
    const int* __restrict__ dst,
    float* __restrict__ out,
    int E)
{
    const int lane  = threadIdx.x & 31;
    const int wave  = blockIdx.x * (BLOCK_THREADS >> 5) + (threadIdx.x >> 5);
    const int m     = lane & 15;        // edge slot within this wave's 16 edges
    const int half  = lane >> 4;        // which 32-float half of the feature dim

    const int e0 = wave * EDGES_PER_WAVE;
    int e = e0 + m;
    // Clamp (branch-free) so every lane issues valid loads; EXEC stays all-1s
    // for the WMMA below. Out-of-range stores are masked at the end.
    int ec = e < E ? e : (E - 1);

    const long long si = (long long)src[ec];
    const long long di = (long long)dst[ec];

    const float* __restrict__ srow = h + si * D_FEAT + half * 32;
    const float* __restrict__ drow = h + di * D_FEAT + half * 32;

    // Each lane accumulates 32 products into 2 partials -> A-matrix slot
    // A[m][K] with K = 2*half + {0,1} (matches 32-bit A 16x4 VGPR layout:
    // lanes 0-15 hold K=0,1; lanes 16-31 hold K=2,3).
    v2f acc; acc.x = 0.0f; acc.y = 0.0f;
#pragma unroll
    for (int c = 0; c < 8; ++c) {
        v4f s = *(const v4f*)(srow + c * 4);   // global_load_b128 (L2-hit gather)
        v4f t = *(const v4f*)(drow + c * 4);
        acc.x = __builtin_fmaf(s.x, t.x, acc.x);
        acc.y = __builtin_fmaf(s.y, t.y, acc.y);
        acc.x = __builtin_fmaf(s.z, t.z, acc.x);
        acc.y = __builtin_fmaf(s.w, t.w, acc.y);
    }

    // Reduce the 4 partials per edge with one matrix op:
    // D(16x16) = A(16x4, partials) x B(4x16, ones)  =>  D[m][j] = score(e0+m)  for all j.
    v2f ones; ones.x = 1.0f; ones.y = 1.0f;
    v8f d = {};
    d = __builtin_amdgcn_wmma_f32_16x16x4_f32(
        /*neg_a=*/false, acc,
        /*neg_b=*/false, ones,
        /*c_mod=*/(short)0, d,
        /*reuse_a=*/false, /*reuse_b=*/false);

    // D layout (32-bit C/D 16x16): lane 0 has D[0..7][0] in d[0..7],
    // lane 16 has D[8..15][0] in d[0..7]. Those two lanes store 8 scores each.
    if ((lane & 15) == 0) {
        const int base = e0 + half * 8;
        if (base + 7 < E) {
            v4f lo; lo.x = d[0]; lo.y = d[1]; lo.z = d[2]; lo.w = d[3];
            v4f hi; hi.x = d[4]; hi.y = d[5]; hi.z = d[6]; hi.w = d[7];
            *(v4f*)(out + base)     = lo;   // global_store_b128
            *(v4f*)(out + base + 4) = hi;   // global_store_b128
        } else {
#pragma unroll
            for (int v = 0; v < 8; ++v) {
                if (base + v < E) out[base + v] = d[v];
            }
        }
    }
}

extern "C" void kernel_launch(void* const* d_in, const int* in_sizes, int n_in,
                              void* d_out, int out_size, void* d_ws, size_t ws_size,
                              hipStream_t stream) {
    const float* h   = (const float*)d_in[0];
    const int*   src = (const int*)d_in[1];   // JAX default x64-disabled -> int32 indices
    const int*   dst = (const int*)d_in[2];
    float*       out = (float*)d_out;

    const int E = in_sizes[1];
    const int blocks = (E + EDGES_PER_BLOCK - 1) / EDGES_PER_BLOCK;

    edge_dot_wmma_kernel<<<blocks, BLOCK_THREADS, 0, stream>>>(h, src, dst, out, E);
}